// RNNEncoder_24404004176037
// MI455X (gfx1250) — compile-verified
//
#include <hip/hip_runtime.h>

typedef __attribute__((ext_vector_type(2))) float v2f;
typedef __attribute__((ext_vector_type(8))) float v8f;

#define HDIM 8

// GRU final-hidden-state kernel, H=8, I=1.
// One wave32 owns 16 batch rows. Recurrent GEMM gh = W_hh(24x8) @ h^T(8x16)
// done with V_WMMA_F32_16X16X4_F32: 2 M-tiles (r/z gates, n gates) x 2 K-slices.
// D layout: lane(0-15)=batch N, vgpr j = gate M=j (lanes0-15) / M=j+8 (lanes16-31).
__global__ __launch_bounds__(256) void gru_wmma_kernel(
    const float* __restrict__ x,
    const float* __restrict__ W_ih,
    const float* __restrict__ W_hh,
    const float* __restrict__ b_ih,
    const float* __restrict__ b_hh,
    float* __restrict__ out,
    int Bsz, int T)
{
  const int tid  = blockIdx.x * blockDim.x + threadIdx.x;
  const int wave = tid >> 5;
  const int lane = tid & 31;
  const int half = lane >> 4;   // 0: lanes 0-15, 1: lanes 16-31
  const int m16  = lane & 15;   // A-matrix row / batch-in-group
  const int b0   = wave * 16;   // first batch row of this wave

  // ---- A tiles (W_hh) in WMMA A layout: lanes0-15 K={0,1}, lanes16-31 K={2,3}
  v2f a0k0, a0k1, a1k0, a1k1;
  {
    const int kb = 2 * half;
    a0k0.x = W_hh[m16 * HDIM + kb + 0];
    a0k0.y = W_hh[m16 * HDIM + kb + 1];
    a0k1.x = W_hh[m16 * HDIM + 4 + kb + 0];
    a0k1.y = W_hh[m16 * HDIM + 4 + kb + 1];
    const bool vr = (m16 < HDIM);          // n-tile valid rows: g = 16+m16, m16<8
    a1k0.x = vr ? W_hh[(16 + m16) * HDIM + kb + 0] : 0.f;
    a1k0.y = vr ? W_hh[(16 + m16) * HDIM + kb + 1] : 0.f;
    a1k1.x = vr ? W_hh[(16 + m16) * HDIM + 4 + kb + 0] : 0.f;
    a1k1.y = vr ? W_hh[(16 + m16) * HDIM + 4 + kb + 1] : 0.f;
  }

  // Per-lane affine constants for the r/z C-tile: gate g = j + 8*half
  float wiA[8], bA[8];
  #pragma unroll
  for (int j = 0; j < 8; ++j) {
    int g = j + 8 * half;
    wiA[j] = W_ih[g];
    bA[j]  = b_ih[g] + b_hh[g];
  }
  // n-gate constants (reference: n = tanh(gi_n + r * (W_hh_n h + b_hh_n)))
  float win[8], bin[8];
  v8f c1;
  #pragma unroll
  for (int j = 0; j < 8; ++j) {
    win[j] = W_ih[16 + j];
    bin[j] = b_ih[16 + j];
    c1[j]  = b_hh[16 + j];
  }

  float h[8];
  #pragma unroll
  for (int j = 0; j < 8; ++j) h[j] = 0.f;

  // B tiles = h^T in WMMA B layout; h0 = 0
  v2f bk0; bk0.x = 0.f; bk0.y = 0.f;
  v2f bk1; bk1.x = 0.f; bk1.y = 0.f;

  const float4* xrow =
      reinterpret_cast<const float4*>(x + (size_t)(b0 + m16) * (size_t)T);

  auto step = [&](float xv) {
    // C0 = x_t*W_ih + (b_ih + b_hh) for r/z gates
    v8f c0;
    #pragma unroll
    for (int j = 0; j < 8; ++j) c0[j] = fmaf(xv, wiA[j], bA[j]);
    // r/z preacts: acc0 = W_hh[0:16,:] @ h + c0   (K=8 via two K=4 slices)
    v8f acc0 = __builtin_amdgcn_wmma_f32_16x16x4_f32(
        false, a0k0, false, bk0, (short)0, c0, false, false);
    acc0 = __builtin_amdgcn_wmma_f32_16x16x4_f32(
        false, a0k1, false, bk1, (short)0, acc0, false, false);
    // n recurrent part: acc1 = W_hh[16:24,:] @ h + b_hh_n
    v8f acc1 = __builtin_amdgcn_wmma_f32_16x16x4_f32(
        false, a1k0, false, bk0, (short)0, c1, false, false);
    acc1 = __builtin_amdgcn_wmma_f32_16x16x4_f32(
        false, a1k1, false, bk1, (short)0, acc1, false, false);

    #pragma unroll
    for (int j = 0; j < 8; ++j) {
      // sigmoid via hardware rcp (v_rcp_f32), not IEEE division
      float s = __builtin_amdgcn_rcpf(1.0f + __expf(-acc0[j]));
      float z = __shfl_xor(s, 16, 32);            // bring z down to r lanes
      float npre = fmaf(xv, win[j], bin[j]) + s * acc1[j];
      // tanh(npre) = 1 - 2/(exp(2*npre)+1), saturation-safe with rcp
      float e = __expf(2.0f * npre);
      float n = fmaf(-2.0f, __builtin_amdgcn_rcpf(e + 1.0f), 1.0f);
      h[j] = fmaf(z, h[j] - n, n);                // (1-z)*n + z*h
    }
    // Rebuild B tiles (h^T): vgpr0 = {K0 | K2}, vgpr1 = {K1 | K3} per K-slice
    float h2s = __shfl_xor(h[2], 16, 32);
    float h3s = __shfl_xor(h[3], 16, 32);
    float h6s = __shfl_xor(h[6], 16, 32);
    float h7s = __shfl_xor(h[7], 16, 32);
    bk0.x = (half == 0) ? h[0] : h2s;
    bk0.y = (half == 0) ? h[1] : h3s;
    bk1.x = (half == 0) ? h[4] : h6s;
    bk1.y = (half == 0) ? h[5] : h7s;
  };

  const int nt4 = T >> 2;
  for (int t4 = 0; t4 < nt4; ++t4) {
    float4 xb = xrow[t4];
    step(xb.x); step(xb.y); step(xb.z); step(xb.w);
  }

  if (half == 0) {
    float4 lo = make_float4(h[0], h[1], h[2], h[3]);
    float4 hi = make_float4(h[4], h[5], h[6], h[7]);
    float4* orow = reinterpret_cast<float4*>(out + (size_t)(b0 + m16) * HDIM);
    orow[0] = lo;
    orow[1] = hi;
  }
}

extern "C" void kernel_launch(void* const* d_in, const int* in_sizes, int n_in,
                              void* d_out, int out_size, void* d_ws, size_t ws_size,
                              hipStream_t stream) {
  const float* x    = (const float*)d_in[0];
  const float* W_ih = (const float*)d_in[1];
  const float* W_hh = (const float*)d_in[2];
  const float* b_ih = (const float*)d_in[3];
  const float* b_hh = (const float*)d_in[4];
  float* out = (float*)d_out;

  const int B = out_size / HDIM;     // out is [B, H]
  const int T = in_sizes[0] / B;     // x is [B, T]

  const int waves   = B / 16;        // 16 batch rows per wave32
  const int threads = 256;           // 8 waves per block
  const int blocks  = (waves * 32 + threads - 1) / threads;

  gru_wmma_kernel<<<blocks, threads, 0, stream>>>(x, W_ih, W_hh, b_ih, b_hh,
                                                  out, B, T);
}